// Retrace_5789615915486
// MI455X (gfx1250) — compile-verified
//
#include <hip/hip_runtime.h>
#include <stdint.h>

#define GAMMA 0.99f
#define TP1   1024          // row length (T+1)
#define TB    256           // threads per block
#define CH    4             // timesteps per thread (TP1 / TB)

typedef unsigned int u32;
typedef u32 u32x4 __attribute__((ext_vector_type(4)));
typedef int  i32x4 __attribute__((ext_vector_type(4)));
typedef int  i32x8 __attribute__((ext_vector_type(8)));

// TDM path only exists in the gfx1250 device pass.
#if defined(__gfx1250__) && __has_builtin(__builtin_amdgcn_tensor_load_to_lds)
#define USE_TDM 1
#else
#define USE_TDM 0
#endif

#if USE_TDM
// Issue one TDM DMA of a contiguous 1024-float row: global -> LDS.
// Descriptor layout per CDNA5 ISA ch.8 (D# groups 0/1; higher-dim groups
// zeroed, tile is 1-D). This toolchain's builtin takes 6 args:
// (u32x4 g0, i32x8 g1, i32x4, i32x4, i32x8, i32 cpol).
__device__ __forceinline__ void tdm_row_load(const float* gsrc, u32 lds_byte_addr) {
  unsigned long long ga = (unsigned long long)(uintptr_t)gsrc;
  u32x4 g0 = {
    1u,                                                  // count=1 (valid), user mode
    lds_byte_addr,                                       // lds_addr [63:32]
    (u32)(ga & 0xFFFFFFFFull),                           // global_addr[31:0]
    ((u32)((ga >> 32) & 0x01FFFFFFull)) | 0x80000000u    // global_addr[56:32] | type=2
  };
  i32x8 g1 = {
    0x00020000,                        // workgroup_mask=0, data_size=2 (4 bytes)
    (int)((TP1 & 0xFFFF) << 16),       // tensor_dim0[15:0] at bits 63:48
    (int)((TP1 >> 16) | (1 << 16)),    // tensor_dim0[31:16], tensor_dim1=1
    (int)((TP1 & 0xFFFF) << 16),       // tensor_dim1 hi=0, tile_dim0=1024
    1,                                 // tile_dim1=1, tile_dim2=0
    (int)TP1,                          // tensor_dim0_stride[31:0]
    0, 0                               // stride hi / dim1_stride (unused, Y runs once)
  };
  i32x4 z4 = {0, 0, 0, 0};             // groups 2/3: dims 3/4 unused
  i32x8 z8 = {0, 0, 0, 0, 0, 0, 0, 0}; // extra descriptor words: unused
  __builtin_amdgcn_tensor_load_to_lds(g0, g1, z4, z4, z8, 0);
}
#endif

__global__ __launch_bounds__(TB)
void retrace_row_kernel(const float* __restrict__ Q,  const float* __restrict__ eQ,
                        const float* __restrict__ tQ, const float* __restrict__ rw,
                        const float* __restrict__ tp, const float* __restrict__ bp,
                        float* __restrict__ partial) {
  __shared__ float smem[6 * TP1 + 2 * TB];
  float* sQ  = smem;
  float* sR  = smem + 1 * TP1;
  float* sEQ = smem + 2 * TP1;
  float* sTQ = smem + 3 * TP1;
  float* sTP = smem + 4 * TP1;
  float* sBP = smem + 5 * TP1;
  float* sA  = smem + 6 * TP1;
  float* sB  = smem + 6 * TP1 + TB;

  const int tid = threadIdx.x;
  const size_t row = (size_t)blockIdx.x * TP1;

#if USE_TDM
  if (tid == 0) {
    // low 32 bits of the flat pointer to LDS == LDS byte offset (ISA 10.2)
    u32 base = (u32)(uintptr_t)&smem[0];
    tdm_row_load(Q  + row, base + 0u * TP1 * 4u);
    tdm_row_load(rw + row, base + 1u * TP1 * 4u);
    tdm_row_load(eQ + row, base + 2u * TP1 * 4u);
    tdm_row_load(tQ + row, base + 3u * TP1 * 4u);
    tdm_row_load(tp + row, base + 4u * TP1 * 4u);
    tdm_row_load(bp + row, base + 5u * TP1 * 4u);
#if __has_builtin(__builtin_amdgcn_s_wait_tensorcnt)
    __builtin_amdgcn_s_wait_tensorcnt(0);
#else
    asm volatile("s_wait_tensorcnt 0x0" ::: "memory");
#endif
  }
  __syncthreads();
#else
  {
    const int i = tid * CH;            // contiguous float4 per array: fully coalesced
    *(float4*)(sQ  + i) = *(const float4*)(Q  + row + i);
    *(float4*)(sR  + i) = *(const float4*)(rw + row + i);
    *(float4*)(sEQ + i) = *(const float4*)(eQ + row + i);
    *(float4*)(sTQ + i) = *(const float4*)(tQ + row + i);
    *(float4*)(sTP + i) = *(const float4*)(tp + row + i);
    *(float4*)(sBP + i) = *(const float4*)(bp + row + i);
  }
  __syncthreads();
#endif

  // Backward recurrence z_j = a_j + b_j * z_{j+1}, valid j = 1..TP1-2,
  // with z_{TP1-1} = init = tQ[TP1-1].
  // Pass 1: build chunk's affine composition, caching (a_j, b_j) in VGPRs.
  // Out-of-range slots (j=0, j=TP1-1) hold identity (0,1).
  float ajr[CH], bjr[CH];
  float A = 0.0f, Bc = 1.0f;
#pragma unroll
  for (int k = CH - 1; k >= 0; --k) {
    const int j = tid * CH + k;
    float aj = 0.0f, bj = 1.0f;
    if (j >= 1 && j <= TP1 - 2) {
      float d = sTP[j + 1] - sBP[j + 1];
      float c = __expf(fminf(d, 0.0f));
      aj = fmaf(GAMMA, fmaf(-c, sTQ[j + 1], sEQ[j + 1]), sR[j]);
      bj = GAMMA * c;
      A  = fmaf(bj, A, aj);
      Bc = bj * Bc;
    }
    ajr[k] = aj;
    bjr[k] = bj;
  }

  // Inclusive suffix scan of affine compositions (Hillis-Steele, 8 rounds).
  sA[tid] = A; sB[tid] = Bc;
  __syncthreads();
  for (int d = 1; d < TB; d <<= 1) {
    float a2 = 0.0f, b2 = 1.0f;
    const bool has = (tid + d) < TB;
    if (has) { a2 = sA[tid + d]; b2 = sB[tid + d]; }
    __syncthreads();
    if (has) { A = fmaf(Bc, a2, A); Bc = Bc * b2; }
    sA[tid] = A; sB[tid] = Bc;
    __syncthreads();
  }

  // Exclusive-suffix carry for this chunk, then replay chunk from registers.
  const float zinit = sTQ[TP1 - 1];
  float z = (tid == TB - 1) ? zinit : fmaf(sB[tid + 1], zinit, sA[tid + 1]);
  float sq = 0.0f;
#pragma unroll
  for (int k = CH - 1; k >= 0; --k) {
    const int j = tid * CH + k;
    z = fmaf(bjr[k], z, ajr[k]);         // z_j (identity slots leave z unchanged)
    if (j >= 1 && j <= TP1 - 2) {
      float diff = sQ[j - 1] - z;
      sq = fmaf(diff, diff, sq);
    }
  }
  if (tid == TB - 1) {                   // j = TP1-1 term: Q_ret last col == init
    float diff = sQ[TP1 - 2] - zinit;
    sq = fmaf(diff, diff, sq);
  }

  // Block reduction (reuse sA after a barrier), deterministic.
  __syncthreads();
  sA[tid] = sq;
  __syncthreads();
  for (int s = TB / 2; s > 0; s >>= 1) {
    if (tid < s) sA[tid] += sA[tid + s];
    __syncthreads();
  }
  if (tid == 0) partial[blockIdx.x] = sA[0];
}

__global__ __launch_bounds__(256)
void retrace_finalize_kernel(const float* __restrict__ partial, int nb,
                             float* __restrict__ out, float inv_n) {
  __shared__ float red[256];
  float s = 0.0f;
  for (int i = threadIdx.x; i < nb; i += 256) s += partial[i];
  red[threadIdx.x] = s;
  __syncthreads();
  for (int st = 128; st > 0; st >>= 1) {
    if (threadIdx.x < st) red[threadIdx.x] += red[threadIdx.x + st];
    __syncthreads();
  }
  if (threadIdx.x == 0) out[0] = red[0] * inv_n;
}

extern "C" void kernel_launch(void* const* d_in, const int* in_sizes, int n_in,
                              void* d_out, int out_size, void* d_ws, size_t ws_size,
                              hipStream_t stream) {
  const float* Q  = (const float*)d_in[0];
  const float* eQ = (const float*)d_in[1];
  const float* tQ = (const float*)d_in[2];
  const float* rw = (const float*)d_in[3];
  const float* tp = (const float*)d_in[4];
  const float* bp = (const float*)d_in[5];
  const int B = in_sizes[0] / TP1;           // 4096 rows
  float* partial = (float*)d_ws;             // B floats of scratch

  retrace_row_kernel<<<B, TB, 0, stream>>>(Q, eQ, tQ, rw, tp, bp, partial);

  const float inv_n = 1.0f / ((float)B * (float)(TP1 - 1));
  retrace_finalize_kernel<<<1, 256, 0, stream>>>(partial, B, (float*)d_out, inv_n);
}